// Scaled_Dot_Product_Attention_2052994367930
// MI455X (gfx1250) — compile-verified
//
#include <hip/hip_runtime.h>
#include <math.h>

typedef float v2f __attribute__((ext_vector_type(2)));
typedef float v8f __attribute__((ext_vector_type(8)));

#define SEQ 2048
#define HD 64
#define MT 32               // query rows per workgroup
#define SROW (SEQ + 4)      // padded LDS row stride; 2052 % 64 == 4 -> bank-conflict-free
#define NTHREADS 256
#define ATT_SCALE 0.125f    // 1/sqrt(64)

__global__ __launch_bounds__(NTHREADS)
void attn_fp32_wmma_kernel(const float* __restrict__ Q,
                           const float* __restrict__ K,
                           const float* __restrict__ V,
                           float* __restrict__ Out,
                           float* __restrict__ Probs) {
  extern __shared__ float smem[];
  float* sc      = smem;                   // MT x SROW score/prob strip
  float* red     = smem + MT * SROW;       // NTHREADS reduce scratch
  float* rowstat = red + NTHREADS;         // MT row stats (max, then 1/sum)

  const int bh   = blockIdx.y;
  const int q0   = blockIdx.x * MT;
  const int tid  = threadIdx.x;
  const int lane = tid & 31;
  const int wave = tid >> 5;
  const int m    = lane & 15;   // row/col within a 16x16 fragment
  const int half = lane >> 4;   // lane group 0/1 (selects K pair in A/B frags)

  const float* Qb = Q + (size_t)bh * SEQ * HD;
  const float* Kb = K + (size_t)bh * SEQ * HD;
  const float* Vb = V + (size_t)bh * SEQ * HD;

  // ---------------- Phase 1: scores = Q * K^T into LDS --------------------
  {
    const int qsub = wave >> 2;            // 0..1 : which 16-row sub-tile
    const int kq   = wave & 3;             // 0..3 : which 512-col span
    // Preload all 16 A-fragments for this wave's 16 Q rows.
    // A 16x4 fp32 layout: lane m holds row m; lanes 0-15 K={4i,4i+1}, lanes 16-31 K={4i+2,4i+3}
    v2f afrag[16];
    const float* qrow = Qb + (size_t)(q0 + qsub * 16 + m) * HD + 2 * half;
#pragma unroll
    for (int i = 0; i < 16; ++i)
      afrag[i] = *(const v2f*)(qrow + 4 * i);

    for (int ct = 0; ct < 32; ++ct) {
      const int n0 = kq * 512 + ct * 16;
      const float* krow = Kb + (size_t)(n0 + m) * HD + 2 * half;  // B[n][k] = K[n][k]
      v8f acc = {};
#pragma unroll
      for (int i = 0; i < 16; ++i) {
        v2f b = *(const v2f*)(krow + 4 * i);
        acc = __builtin_amdgcn_wmma_f32_16x16x4_f32(
            false, afrag[i], false, b, (short)0, acc, false, false);
      }
      // C/D layout: VGPR v -> row v + 8*half, col = m
      float* dst = sc + (size_t)(qsub * 16 + 8 * half) * SROW + n0 + m;
#pragma unroll
      for (int v = 0; v < 8; ++v)
        dst[(size_t)v * SROW] = acc[v];
    }
  }
  __syncthreads();

  // ---------------- Phase 2: softmax rows + stream probs to global --------
  {
    const int row = tid >> 3;              // 8 threads per row
    const int c0  = tid & 7;
    float* r = sc + (size_t)row * SROW;

    float mx = -3.402823466e38f;
    for (int i = c0; i < SEQ; i += 8) mx = fmaxf(mx, r[i]);
    red[tid] = mx;
    __syncthreads();
    if (tid < MT) {
      float v = red[tid * 8];
      for (int j = 1; j < 8; ++j) v = fmaxf(v, red[tid * 8 + j]);
      rowstat[tid] = v;
    }
    __syncthreads();

    const float rm = rowstat[row] * ATT_SCALE;
    float sum = 0.f;
    for (int i = c0; i < SEQ; i += 8) {
      float e = __expf(r[i] * ATT_SCALE - rm);
      r[i] = e;
      sum += e;
    }
    red[tid] = sum;
    __syncthreads();
    if (tid < MT) {
      float v = 0.f;
      for (int j = 0; j < 8; ++j) v += red[tid * 8 + j];
      rowstat[tid] = 1.0f / v;
    }
    __syncthreads();

    // Normalize in LDS and stream probs to global with NT stores (write-once
    // 402MB stream must not evict the L2-resident Q/K/V working set).
    float* pout = Probs + ((size_t)bh * SEQ + q0) * SEQ;
    for (int idx = tid; idx < MT * SEQ; idx += NTHREADS) {
      const int rr = idx >> 11;            // / 2048
      const int cc = idx & (SEQ - 1);
      const float p = sc[(size_t)rr * SROW + cc] * rowstat[rr];
      sc[(size_t)rr * SROW + cc] = p;
      __builtin_nontemporal_store(p, &pout[(size_t)rr * SEQ + cc]);
    }
  }
  __syncthreads();

  // ---------------- Phase 3: out = P * V ----------------------------------
  // 4 independent WMMA accumulation chains to cover WMMA latency (single
  // output tile per wave would otherwise serialize 512 dependent wmmas).
  {
    const int qsub = wave >> 2;            // 0..1
    const int n0   = (wave & 3) * 16;      // 0,16,32,48 within D=64
    v8f acc0 = {}, acc1 = {}, acc2 = {}, acc3 = {};
    const float* sp = sc + (size_t)(qsub * 16 + m) * SROW + 2 * half;   // A from LDS probs
    const float* vp = Vb + (size_t)(2 * half) * HD + n0 + m;            // B from V
    for (int k0 = 0; k0 < SEQ; k0 += 16) {
      v2f a0 = *(const v2f*)(sp + k0);
      v2f a1 = *(const v2f*)(sp + k0 + 4);
      v2f a2 = *(const v2f*)(sp + k0 + 8);
      v2f a3 = *(const v2f*)(sp + k0 + 12);
      v2f b0, b1, b2, b3;
      b0.x = vp[(size_t)(k0     ) * HD]; b0.y = vp[(size_t)(k0     ) * HD + HD];
      b1.x = vp[(size_t)(k0 +  4) * HD]; b1.y = vp[(size_t)(k0 +  4) * HD + HD];
      b2.x = vp[(size_t)(k0 +  8) * HD]; b2.y = vp[(size_t)(k0 +  8) * HD + HD];
      b3.x = vp[(size_t)(k0 + 12) * HD]; b3.y = vp[(size_t)(k0 + 12) * HD + HD];
      acc0 = __builtin_amdgcn_wmma_f32_16x16x4_f32(false, a0, false, b0, (short)0, acc0, false, false);
      acc1 = __builtin_amdgcn_wmma_f32_16x16x4_f32(false, a1, false, b1, (short)0, acc1, false, false);
      acc2 = __builtin_amdgcn_wmma_f32_16x16x4_f32(false, a2, false, b2, (short)0, acc2, false, false);
      acc3 = __builtin_amdgcn_wmma_f32_16x16x4_f32(false, a3, false, b3, (short)0, acc3, false, false);
    }
    v8f acc = (acc0 + acc1) + (acc2 + acc3);
    float* orow = Out + ((size_t)bh * SEQ + q0 + qsub * 16 + 8 * half) * HD + n0 + m;
#pragma unroll
    for (int v = 0; v < 8; ++v)
      orow[(size_t)v * HD] = acc[v];
  }
}

extern "C" void kernel_launch(void* const* d_in, const int* in_sizes, int n_in,
                              void* d_out, int out_size, void* d_ws, size_t ws_size,
                              hipStream_t stream) {
  (void)in_sizes; (void)n_in; (void)out_size; (void)d_ws; (void)ws_size;
  const float* Q = (const float*)d_in[0];
  const float* K = (const float*)d_in[1];
  const float* V = (const float*)d_in[2];
  float* out   = (float*)d_out;                                // [B,H,S,D]
  float* probs = out + (size_t)2 * 12 * SEQ * HD;              // [B,H,S,S]

  const size_t shmem = ((size_t)MT * SROW + NTHREADS + MT) * sizeof(float); // ~258 KB
  // Best-effort opt-in for >64KB dynamic LDS (ignored/no-op on ROCm if unsupported).
  (void)hipFuncSetAttribute((const void*)attn_fp32_wmma_kernel,
                            hipFuncAttributeMaxDynamicSharedMemorySize, (int)shmem);

  dim3 grid(SEQ / MT, 2 * 12);
  attn_fp32_wmma_kernel<<<grid, NTHREADS, shmem, stream>>>(Q, K, V, out, probs);
}